// UNCL_31490700214618
// MI455X (gfx1250) — compile-verified
//
#include <hip/hip_runtime.h>
#include <hip/hip_bf16.h>

#define N_PTS   1024
#define DIM     256
#define NCLS    10
#define EPSF    1e-12f

#define TI      32
#define TJ      64
#define DCHUNK  128   // stage D in 2 chunks -> ~129KB LDS -> 2 workgroups/WGP

typedef __attribute__((ext_vector_type(2))) float v2f;
typedef __attribute__((ext_vector_type(8))) float v8f;

#define WS_INTS 6656   // 26624 bytes zeroed

__global__ void k_init(int* ws, float* out) {
    for (int e = threadIdx.x; e < WS_INTS; e += blockDim.x) ws[e] = 0;
    if (threadIdx.x == 0) out[0] = 0.0f;
}

__global__ void k_hist(const int* __restrict__ tgt, int* cnt) {
    int i = blockIdx.x * blockDim.x + threadIdx.x;
    if (i < N_PTS) atomicAdd(&cnt[tgt[i]], 1);
}

__global__ void k_qtab(const int* __restrict__ cnt, float* qp, float* qm) {
    int idx = threadIdx.x;
    if (idx < NCLS * NCLS) {
        int ci = idx / NCLS, cj = idx % NCLS;
        float np = (float)cnt[ci];
        float nn = (float)N_PTS - np;
        float inv2N = 0.5f / (float)N_PTS;
        if (ci == cj) { qp[idx] = inv2N / np; qm[idx] = inv2N / np; }
        else          { qp[idx] = inv2N / nn; qm[idx] = -inv2N / nn; }
    }
}

__global__ void k_tcls(const float* __restrict__ fm_t, const int* __restrict__ tgt,
                       float* tc1, float* tc2) {
    int j = blockIdx.x * blockDim.x + threadIdx.x;
    if (j >= N_PTS) return;
    int c = tgt[j];
    for (int d = 0; d < DIM; ++d) {
        float t = fm_t[j * DIM + d];
        atomicAdd(&tc1[c * DIM + d], t);
        atomicAdd(&tc2[c * DIM + d], t * t);
    }
}

__global__ void k_spart(const float* __restrict__ fm_s, const float* __restrict__ lv_s,
                        const int* __restrict__ tgt, const float* __restrict__ qm,
                        const float* __restrict__ tc1, const float* __restrict__ tc2,
                        float* out) {
    int i = blockIdx.x * blockDim.x + threadIdx.x;
    if (i >= N_PTS) return;
    int ci = tgt[i];
    float qrow[NCLS];
#pragma unroll
    for (int c = 0; c < NCLS; ++c) qrow[c] = qm[ci * NCLS + c];
    float term = 0.0f;
    for (int d = 0; d < DIM; ++d) {
        float lv = lv_s[i * DIM + d];
        float s  = fm_s[i * DIM + d];
        float a  = 1.0f / (2.0f * __expf(lv) + EPSF);
        float b  = 2.0f * a * s;
        float u1 = 0.0f, u2 = 0.0f;
#pragma unroll
        for (int c = 0; c < NCLS; ++c) {
            u1 += qrow[c] * tc1[c * DIM + d];
            u2 += qrow[c] * tc2[c * DIM + d];
        }
        term += a * u2 - b * u1;
    }
    atomicAdd(out, term);
}

// rank-3 outer product via WMMA:  X[m,n] = c - b*t + a*t^2  for one d.
__device__ __forceinline__ v8f wmma_rank3(v2f a2, v2f b2) {
    v8f z = {0.f, 0.f, 0.f, 0.f, 0.f, 0.f, 0.f, 0.f};
    return __builtin_amdgcn_wmma_f32_16x16x4_f32(
        false, a2, false, b2, (short)0, z, false, false);
}

// |X| accumulate with VOP3 abs source modifier (1 VALU per element).
__device__ __forceinline__ void abs_acc(const v8f& x, float acc[8]) {
#pragma unroll
    for (int r = 0; r < 8; ++r)
        asm("v_add_f32 %0, %0, |%1|" : "+v"(acc[r]) : "v"(x[r]));
}

__global__ void k_main(const float* __restrict__ fm_s, const float* __restrict__ fm_t,
                       const float* __restrict__ lv_s, const int* __restrict__ tgt,
                       const float* __restrict__ qp, float* out) {
    extern __shared__ char smem[];
    float2* SA = (float2*)smem;            // [TI][DCHUNK]  (b, a)     A K0,K1  lanes 0-15
    float2* SC = SA + TI * DCHUNK;         // [TI][DCHUNK]  (c, 0)     A K2,K3  lanes 16-31
    float2* TB = SC + TI * DCHUNK;         // [TJ][DCHUNK]  (-t, t^2)  B K0,K1  lanes 0-15
    float2* CB = TB + TJ * DCHUNK;         // [DCHUNK]      (1, 0)     B K2,K3  lanes 16-31

    const int tid = threadIdx.x;
    const int i0 = blockIdx.x * TI;
    const int j0 = blockIdx.y * TJ;

    const int lane = tid & 31;
    const int m    = lane & 15;
    const int hi   = lane >> 4;
    const int wave = tid >> 5;
    const int iloc = (wave >> 2) * 16;
    const int jloc = (wave & 3) * 16;

    const float2* baseA = hi ? (SC + (iloc + m) * DCHUNK) : (SA + (iloc + m) * DCHUNK);
    const float2* baseB = hi ? CB : (TB + (jloc + m) * DCHUNK);

    float acc[8];
#pragma unroll
    for (int r = 0; r < 8; ++r) acc[r] = 0.0f;

    for (int dc = 0; dc < DIM; dc += DCHUNK) {
        __syncthreads();   // previous chunk fully consumed before refill
        for (int e = tid; e < TI * DCHUNK; e += 256) {
            int i = e >> 7, d = e & (DCHUNK - 1);
            int gi = i0 + i;
            float lv = lv_s[gi * DIM + dc + d];
            float s  = fm_s[gi * DIM + dc + d];
            float a  = 1.0f / (2.0f * __expf(lv) + EPSF);
            float2 ba; ba.x = 2.0f * a * s; ba.y = a;
            SA[e] = ba;
            float2 cz; cz.x = a * s * s + 0.5f * lv; cz.y = 0.0f;
            SC[e] = cz;
        }
        for (int e = tid; e < TJ * DCHUNK; e += 256) {
            int j = e >> 7, d = e & (DCHUNK - 1);
            float t = fm_t[(j0 + j) * DIM + dc + d];
            float2 tt; tt.x = -t; tt.y = t * t;
            TB[e] = tt;
        }
        if (tid < DCHUNK) { float2 one; one.x = 1.0f; one.y = 0.0f; CB[tid] = one; }
        __syncthreads();

#pragma unroll 8
        for (int d = 0; d < DCHUNK; d += 2) {
            float4 a4 = *(const float4*)(baseA + d);
            float4 b4 = *(const float4*)(baseB + d);
            v2f a0 = {a4.x, a4.y}, b0 = {b4.x, b4.y};
            v2f a1 = {a4.z, a4.w}, b1 = {b4.z, b4.w};
            v8f x0 = wmma_rank3(a0, b0);
            v8f x1 = wmma_rank3(a1, b1);
            abs_acc(x0, acc);
            abs_acc(x1, acc);
        }
    }

    // acc[r] is element (M = r + 8*hi, N = lane&15) of the 16x16 tile
    int jg = j0 + jloc + m;
    int cj = tgt[jg];
    float partial = 0.0f;
#pragma unroll
    for (int r = 0; r < 8; ++r) {
        int ig = i0 + iloc + r + (hi ? 8 : 0);
        int ci = tgt[ig];
        partial += qp[ci * NCLS + cj] * acc[r];
    }
#pragma unroll
    for (int off = 16; off > 0; off >>= 1)
        partial += __shfl_xor(partial, off, 32);
    if (lane == 0) atomicAdd(out, partial);
}

extern "C" void kernel_launch(void* const* d_in, const int* in_sizes, int n_in,
                              void* d_out, int out_size, void* d_ws, size_t ws_size,
                              hipStream_t stream) {
    const float* fm_s = (const float*)d_in[0];
    const float* fm_t = (const float*)d_in[1];
    const float* lv_s = (const float*)d_in[2];
    const int*   tgt  = (const int*)d_in[3];
    float* out = (float*)d_out;

    char* ws = (char*)d_ws;
    int*   cnt = (int*)ws;
    float* qp  = (float*)(ws + 256);
    float* qm  = (float*)(ws + 1024);
    float* tc1 = (float*)(ws + 4096);
    float* tc2 = (float*)(ws + 16384);

    const size_t smem_bytes =
        (size_t)(TI * DCHUNK + TI * DCHUNK + TJ * DCHUNK + DCHUNK) * sizeof(float2);
    (void)hipFuncSetAttribute((const void*)k_main,
                              hipFuncAttributeMaxDynamicSharedMemorySize,
                              (int)smem_bytes);

    k_init<<<1, 256, 0, stream>>>((int*)ws, out);
    k_hist<<<N_PTS / 256, 256, 0, stream>>>(tgt, cnt);
    k_qtab<<<1, 128, 0, stream>>>(cnt, qp, qm);
    k_tcls<<<N_PTS / 256, 256, 0, stream>>>(fm_t, tgt, tc1, tc2);
    k_spart<<<N_PTS / 256, 256, 0, stream>>>(fm_s, lv_s, tgt, qm, tc1, tc2, out);

    dim3 grid(N_PTS / TI, N_PTS / TJ);
    k_main<<<grid, 256, smem_bytes, stream>>>(fm_s, fm_t, lv_s, tgt, qp, out);
}